// CustomCRF_73529840107983
// MI455X (gfx1250) — compile-verified
//
#include <hip/hip_runtime.h>
#include <stdint.h>

// Batched Viterbi decode (CRF inference): B=256, T=512, C=128.
// Max-plus DP is not expressible on WMMA (sum-product only), so this is a
// VALU+LDS kernel. CDNA5-specific paths used:
//   * tensor_load_to_lds (TDM DMA of the 64KB transition matrix into LDS)
//   * s_wait_tensorcnt
//   * global_prefetch_b8 (__builtin_prefetch of next emission row)
//   * 320KB LDS per WGP: backpointers (511x128 bytes) kept entirely in LDS,
//     so backtracking never touches HBM.

#define CRF_B 256
#define CRF_T 512
#define CRF_C 128

typedef unsigned int v4u __attribute__((ext_vector_type(4)));
typedef int          v8i __attribute__((ext_vector_type(8)));
typedef int          v4i __attribute__((ext_vector_type(4)));

__global__ __launch_bounds__(CRF_C, 1)
void viterbi_crf_kernel(const float* __restrict__ em,      // [B,T,C]
                        const float* __restrict__ trans,   // [C,C]
                        int*         __restrict__ path_out,  // [B,T]
                        float*       __restrict__ score_out) // [B]
{
    __shared__ float         trans_s[CRF_C * CRF_C];        // 64 KB
    __shared__ unsigned char bp_s[(CRF_T - 1) * CRF_C];     // 63.875 KB
    __shared__ float         prev_s[2][CRF_C];              // ping-pong scores
    __shared__ int           path_s[CRF_T];

    const int j = threadIdx.x;           // state index owned by this thread
    const int b = blockIdx.x;            // batch element owned by this block
    const float* em_b = em + (size_t)b * CRF_T * CRF_C;

    // ---- TDM: DMA trans[128][128] (64KB, contiguous) into LDS ----
    // Issue from the first wave; instruction is idempotent if it ends up
    // EXEC-masked in other waves (same src -> same dst).
    if (j < 32) {
        uint64_t ga = (uint64_t)(uintptr_t)trans;
        uint32_t lds_off = (uint32_t)(uintptr_t)trans_s; // low32 of flat = LDS offset
        v4u g0;
        g0[0] = 1u;                                          // count=1, user desc
        g0[1] = lds_off;                                     // lds_addr (bytes)
        g0[2] = (uint32_t)(ga & 0xFFFFFFFFu);                // global_addr lo
        g0[3] = (uint32_t)((ga >> 32) & 0x1FFFFFFu)          // global_addr hi (57b)
              | (2u << 30);                                  // type = 2 ("image")
        v8i g1;
        g1[0] = (2 << 16);        // workgroup_mask=0 (not in cluster), data_size=4B
        g1[1] = (0x4000 << 16);   // tensor_dim0 = 16384 elements (lo16)
        g1[2] = 0;                // tensor_dim0 hi16 = 0; tensor_dim1 = 0
        g1[3] = (0x4000 << 16);   // tensor_dim1 hi16 = 0; tile_dim0 = 16384
        g1[4] = 0;                // tile_dim1/2 = 0 (unused -> 1D)
        g1[5] = 0x4000;           // tensor_dim0_stride = 16384 (lo32)
        g1[6] = 0;                // stride hi / dim1 stride
        g1[7] = 0;
        v4i gz4 = {0, 0, 0, 0};            // groups 2/3 unused (<=2D tensor)
        v8i gz8 = {0, 0, 0, 0, 0, 0, 0, 0}; // extra group (unused state)
        __builtin_amdgcn_tensor_load_to_lds(g0, g1, gz4, gz4, gz8, /*cpol=*/0);
    }

    // Initial scores: prev = emissions[b, 0, :]
    prev_s[0][j] = em_b[j];

    __builtin_amdgcn_s_wait_tensorcnt(0);   // TENSORcnt==0 (no-op for non-issuers)
    __syncthreads();                        // trans_s + prev_s visible to all

    // ---- Forward max-plus DP over time ----
    int cur = 0;
    const float* col = &trans_s[j];         // column j of trans, stride C dwords
    for (int t = 1; t < CRF_T; ++t) {
        const float emv = em_b[(size_t)t * CRF_C + j];   // coalesced row load
        if (t + 1 < CRF_T)
            __builtin_prefetch(em_b + (size_t)(t + 1) * CRF_C + j, 0, 1);

        const float* pv = prev_s[cur];
        float best = -__builtin_inff();
        int   bidx = 0;
        // max_i (prev[i] + trans[i][j]); strict '>' in ascending i matches
        // jnp.argmax first-occurrence tie-break.
        #pragma unroll 8
        for (int i0 = 0; i0 < CRF_C; i0 += 4) {
            const float4 p = *reinterpret_cast<const float4*>(pv + i0); // broadcast
            float c0 = p.x + col[(i0 + 0) * CRF_C];
            float c1 = p.y + col[(i0 + 1) * CRF_C];
            float c2 = p.z + col[(i0 + 2) * CRF_C];
            float c3 = p.w + col[(i0 + 3) * CRF_C];
            if (c0 > best) { best = c0; bidx = i0 + 0; }
            if (c1 > best) { best = c1; bidx = i0 + 1; }
            if (c2 > best) { best = c2; bidx = i0 + 2; }
            if (c3 > best) { best = c3; bidx = i0 + 3; }
        }

        bp_s[(size_t)(t - 1) * CRF_C + j] = (unsigned char)bidx;
        prev_s[cur ^ 1][j] = best + emv;
        __syncthreads();                    // one barrier per step (ping-pong)
        cur ^= 1;
    }

    // ---- Backtrack entirely in LDS ----
    if (j == 0) {
        const float* last = prev_s[cur];
        float bestv = last[0];
        int   btag  = 0;
        for (int jj = 1; jj < CRF_C; ++jj) {
            float v = last[jj];
            if (v > bestv) { bestv = v; btag = jj; }     // first-max tie-break
        }
        score_out[b] = bestv;
        path_s[CRF_T - 1] = btag;
        int tag = btag;
        for (int t = CRF_T - 2; t >= 0; --t) {
            tag = (int)bp_s[(size_t)t * CRF_C + tag];
            path_s[t] = tag;
        }
    }
    __syncthreads();

    // Coalesced path writeback: path[b][t], int32
    for (int t = j; t < CRF_T; t += CRF_C)
        path_out[(size_t)b * CRF_T + t] = path_s[t];
}

extern "C" void kernel_launch(void* const* d_in, const int* in_sizes, int n_in,
                              void* d_out, int out_size, void* d_ws, size_t ws_size,
                              hipStream_t stream) {
    (void)in_sizes; (void)n_in; (void)d_ws; (void)ws_size; (void)out_size;
    const float* em    = (const float*)d_in[0];   // [B,T,C] f32
    const float* trans = (const float*)d_in[1];   // [C,C]   f32
    // d_out: path [B,T] int32 followed by final_score [B] f32 (flat concat)
    int*   path_out  = (int*)d_out;
    float* score_out = (float*)d_out + (size_t)CRF_B * CRF_T;

    viterbi_crf_kernel<<<dim3(CRF_B), dim3(CRF_C), 0, stream>>>(
        em, trans, path_out, score_out);
}